// Graph_Learn_27762668601618
// MI455X (gfx1250) — compile-verified
//
#include <hip/hip_runtime.h>
#include <hip/hip_bf16.h>

#define NB 4
#define TB 12
#define VB 200
#define FB 64
#define VT 8          // v-tiles per (n,t) in pass 1 (200 / 25)

typedef __attribute__((ext_vector_type(2))) float v2f;
typedef __attribute__((ext_vector_type(4))) float v4f;
typedef __attribute__((ext_vector_type(8))) float v8f;

// the async-LDS builtins take pointers to this exact type (per hipcc diag)
typedef int v4i __attribute__((vector_size(4 * sizeof(int))));
typedef __attribute__((address_space(1))) v4i* gv4i_p;
typedef __attribute__((address_space(3))) v4i* lv4i_p;

#if __has_builtin(__builtin_amdgcn_global_load_async_to_lds_b128)
#define HAVE_ASYNC_LDS 1
#else
#define HAVE_ASYNC_LDS 0
#endif

#if __has_builtin(__builtin_amdgcn_wmma_f32_16x16x4_f32)
#define HAVE_WMMA4 1
#else
#define HAVE_WMMA4 0
#endif

// ---------------------------------------------------------------------------
// Pass 1: per (n,t): stage x[n,t,:,:] (200x64 f32 = 51.2KB) into LDS via the
// CDNA5 async global->LDS path, compute s[v,w] = sum_f |x_v - x_w| * a_f with
// packed f32 VALU, apply exp, and scatter into d_out with the (V,N,V)->(N,V,V)
// reinterpretation baked into the destination index.
// ---------------------------------------------------------------------------
__global__ __launch_bounds__(256)
void gl_pass1(const float* __restrict__ x, const float* __restrict__ a,
              float* __restrict__ out) {
    __shared__ float xs[VB * FB];   // 51200 B
    __shared__ float as[FB];        // 256 B

    const int blk = blockIdx.x;     // (n*TB + t)*VT + vt
    const int vt  = blk % VT;
    const int nt  = blk / VT;       // n*TB + t
    const int n   = nt / TB;
    const int t   = nt - n * TB;
    const int tid = threadIdx.x;

    const float* xbase = x + (size_t)nt * (VB * FB);

#if HAVE_ASYNC_LDS
    // async DMA-style copy, tracked by ASYNCcnt
    for (int i = tid; i < (VB * FB) / 4; i += 256) {
        __builtin_amdgcn_global_load_async_to_lds_b128(
            (gv4i_p)(xbase + i * 4), (lv4i_p)(&xs[i * 4]), 0, 0);
    }
    if (tid < FB / 4) {
        __builtin_amdgcn_global_load_async_to_lds_b128(
            (gv4i_p)(a + tid * 4), (lv4i_p)(&as[tid * 4]), 0, 0);
    }
#if __has_builtin(__builtin_amdgcn_s_wait_asynccnt)
    __builtin_amdgcn_s_wait_asynccnt(0);
#else
    asm volatile("s_wait_asynccnt 0" ::: "memory");
#endif
#else
    for (int i = tid; i < (VB * FB) / 4; i += 256)
        *(v4f*)&xs[i * 4] = *(const v4f*)(xbase + i * 4);
    if (tid < FB / 4)
        *(v4f*)&as[tid * 4] = *(const v4f*)(a + tid * 4);
#endif
    __syncthreads();

    // 25 v-rows x 200 w-cols = 5000 elements for this block
    for (int e = tid; e < 25 * VB; e += 256) {
        const int vi = e / VB;
        const int w  = e - vi * VB;
        const int v  = vt * 25 + vi;

        const v4f* xv = (const v4f*)&xs[v * FB];
        const v4f* xw = (const v4f*)&xs[w * FB];
        const v4f* av = (const v4f*)as;

        float acc = 0.0f;
#pragma unroll
        for (int f = 0; f < FB / 4; ++f) {
            v4f dv = xv[f], dw = xw[f], aa = av[f];
            acc += __builtin_fabsf(dv.x - dw.x) * aa.x;
            acc += __builtin_fabsf(dv.y - dw.y) * aa.y;
            acc += __builtin_fabsf(dv.z - dw.z) * aa.z;
            acc += __builtin_fabsf(dv.w - dw.w) * aa.w;
        }
        const float ev = __expf(acc);

        // raw (V,N,V)->(N,V,V) reinterpretation
        const int k     = v * NB + n;
        const int n_out = k / VB;
        const int v_out = k - n_out * VB;
        out[(((size_t)n_out * TB + t) * VB + v_out) * VB + w] = ev;
    }
}

// ---------------------------------------------------------------------------
// Pass 2: per (n_out,t): column sums of the 200x200 exp matrix on the MATRIX
// pipe: V_WMMA_F32_16X16X4_F32 with A = 16x4 ones reduces 4 rows x 16 cols per
// issue; accumulate 50 K-steps in the f32 accumulator, then rescale in place.
// One wave owns a 16-column group; 8 waves cover 13 groups.
//
// Out-of-range columns are handled branchlessly: WMMA column n of D depends
// only on column n of B, so invalid lanes just load a clamped (duplicate)
// column and simply never store — no zero-masking, no exec divergence in the
// load/WMMA loop.
// ---------------------------------------------------------------------------
__global__ __launch_bounds__(256)
void gl_pass2(float* __restrict__ out) {
    const int nt   = blockIdx.x;             // n_out*TB + t
    float* M       = out + (size_t)nt * (VB * VB);
    const int wave = threadIdx.x >> 5;
    const int lane = threadIdx.x & 31;
    const int nsub = lane & 15;              // column within group
    const int kh   = lane >> 4;              // K half (0: k=0,1 ; 1: k=2,3)

    for (int g = wave; g < (VB + 15) / 16; g += 8) {
        const int w      = g * 16 + nsub;
        const bool valid = (w < VB);
        const int  wc    = valid ? w : (VB - 1);   // branchless clamp
        const float* Mw  = M + wc;

#if HAVE_WMMA4
        v8f c = {};
        v2f ones; ones.x = 1.0f; ones.y = 1.0f;
#pragma unroll 5
        for (int r = 0; r < VB; r += 4) {
            v2f b;
            b.x = Mw[(r + 2 * kh    ) * VB];
            b.y = Mw[(r + 2 * kh + 1) * VB];
            // D = ones(16x4) x B(4x16) + C  -> every row of D = column sums
            c = __builtin_amdgcn_wmma_f32_16x16x4_f32(
                    false, ones, false, b, (short)0, c, false, false);
        }
        const float colsum = c[0];           // row 0 (and row 8) of D
#else
        float colsum = 0.0f;
        for (int r = 0; r < VB; ++r)
            colsum += Mw[r * VB];
#endif
        const float rs = 1.0f / colsum;

        if (valid) {
            // split the 200 rows across the two K-halves of the wave
            for (int v = kh * 100; v < kh * 100 + 100; ++v)
                M[v * VB + w] = M[v * VB + w] * rs;
        }
    }
}

extern "C" void kernel_launch(void* const* d_in, const int* in_sizes, int n_in,
                              void* d_out, int out_size, void* d_ws, size_t ws_size,
                              hipStream_t stream) {
    const float* x = (const float*)d_in[0];   // (4,12,200,64) f32
    const float* a = (const float*)d_in[1];   // (64,1) f32
    float* out     = (float*)d_out;           // (4,12,200,200) f32
    (void)d_ws; (void)ws_size; (void)in_sizes; (void)n_in; (void)out_size;

    gl_pass1<<<NB * TB * VT, 256, 0, stream>>>(x, a, out);
    gl_pass2<<<NB * TB, 256, 0, stream>>>(out);
}